// ReservoirComputingLayer_29394756174340
// MI455X (gfx1250) — compile-verified
//
#include <hip/hip_runtime.h>
#include <cmath>

// Problem constants (from reference)
#define BATCH   128
#define SEQ     256
#define DIN     512
#define NRES    1024
#define DOUTC   2048   // output row length (padded)
#define WROW    2048   // row stride of reservoir_weights (padded 2048x2048)
#define NWG     64     // one WG per 16-column tile: 64*16 = 1024
#define TPB     256    // 8 waves * 16 rows = 128 batch rows

typedef __attribute__((ext_vector_type(2))) float v2f;
typedef __attribute__((ext_vector_type(4))) float v4f;
typedef __attribute__((ext_vector_type(8))) float v8f;

// Zero the padding columns out[:, :, 1024:2048] and the grid-barrier counter.
__global__ void reservoir_init_kernel(float* __restrict__ out, unsigned* __restrict__ bar) {
    size_t i = (size_t)blockIdx.x * blockDim.x + threadIdx.x;
    if (i == 0) *bar = 0u;
    const size_t total4 = (size_t)BATCH * SEQ * (NRES / 4); // float4 count in pad region
    const size_t stride = (size_t)gridDim.x * blockDim.x;
    v4f z = {0.f, 0.f, 0.f, 0.f};
    for (size_t j = i; j < total4; j += stride) {
        size_t bt = j / (NRES / 4);
        size_t c4 = j % (NRES / 4);
        *(v4f*)(out + bt * DOUTC + NRES + c4 * 4) = z;
    }
}

// Persistent recurrence kernel: 64 WGs x 8 waves = 512 (16x16) tiles of the
// (128 x 1024) state. Per step: pre = x_t @ Win^T + s_{t-1} @ Wres via
// v_wmma_f32_16x16x4_f32; leaky tanh + threshold; state is stored in-place
// as out[:, t, :1024]. Device-wide barrier between steps.
extern "C" __global__ void __launch_bounds__(TPB, 1)
reservoir_step_kernel(const float* __restrict__ x,      // (128, 256, 512)
                      const float* __restrict__ wres,   // (2048, 2048), valid 1024x1024
                      const float* __restrict__ win,    // (2048, 512),  valid 1024x512
                      float* __restrict__ out,          // (128, 256, 2048)
                      unsigned* __restrict__ bar) {
    extern __shared__ char smem[];
    v2f* ldsB   = (v2f*)smem;                          // Wres panel: 512 k-pairs x 16 n = 64KB
    v2f* ldsBin = (v2f*)(smem + 8192 * sizeof(v2f));   // Win^T panel: 256 k-pairs x 16 n = 32KB

    const int tid  = threadIdx.x;
    const int n0   = blockIdx.x * 16;      // this WG's column tile
    const int wave = tid >> 5;
    const int lane = tid & 31;
    const int lm   = lane & 15;            // N index (B/C/D) and M index (A)
    const int hi   = lane >> 4;            // half-wave select
    const int m0   = wave * 16;            // this wave's row tile

    // ---- Stage B panels into LDS in WMMA B layout (once; weights are step-invariant).
    // B frag for k-pair q at column n lives at lds[q*16 + n] = {B[2q][n], B[2q+1][n]}.
    for (int i = tid; i < 8192; i += TPB) {           // Wres: B[k][n] = wres[k*2048 + n0+n]
        int p = i >> 4, n = i & 15;
        v2f v;
        v.x = wres[(size_t)(2 * p)     * WROW + n0 + n];
        v.y = wres[(size_t)(2 * p + 1) * WROW + n0 + n];
        ldsB[i] = v;
    }
    for (int i = tid; i < 4096; i += TPB) {           // Win^T: B[k][n] = win[(n0+n)*512 + k]
        int p = i >> 4, n = i & 15;
        ldsBin[i] = *(const v2f*)(win + (size_t)(n0 + n) * DIN + 2 * p);
    }
    __syncthreads();

    // A-frag base for the input projection: row m0+lm of x, k offset 2*hi.
    const float* aInBase = x + (size_t)(m0 + lm) * SEQ * DIN + 2 * hi;

    for (int t = 0; t < SEQ; ++t) {
        v8f acc0 = {0.f, 0.f, 0.f, 0.f, 0.f, 0.f, 0.f, 0.f};
        v8f acc1 = {0.f, 0.f, 0.f, 0.f, 0.f, 0.f, 0.f, 0.f};

        // ---- input projection: K = 512, 128 WMMAs (dual accumulators)
        const float* aIn = aInBase + (size_t)t * DIN;
        #pragma unroll 8
        for (int i = 0; i < DIN / 4; i += 2) {
            v2f a0 = *(const v2f*)(aIn + 4 * i);
            v2f b0 = ldsBin[(2 * i + hi) * 16 + lm];
            acc0 = __builtin_amdgcn_wmma_f32_16x16x4_f32(false, a0, false, b0,
                                                         (short)0, acc0, false, false);
            v2f a1 = *(const v2f*)(aIn + 4 * (i + 1));
            v2f b1 = ldsBin[(2 * (i + 1) + hi) * 16 + lm];
            acc1 = __builtin_amdgcn_wmma_f32_16x16x4_f32(false, a1, false, b1,
                                                         (short)0, acc1, false, false);
        }

        // ---- recurrent projection: K = 1024, 256 WMMAs; s_{t-1} = out[:, t-1, :1024]
        if (t > 0) {
            const float* aS = out + ((size_t)(m0 + lm) * SEQ + (t - 1)) * DOUTC + 2 * hi;
            #pragma unroll 8
            for (int i = 0; i < NRES / 4; i += 2) {
                v2f a0 = *(const v2f*)(aS + 4 * i);
                v2f b0 = ldsB[(2 * i + hi) * 16 + lm];
                acc0 = __builtin_amdgcn_wmma_f32_16x16x4_f32(false, a0, false, b0,
                                                             (short)0, acc0, false, false);
                v2f a1 = *(const v2f*)(aS + 4 * (i + 1));
                v2f b1 = ldsB[(2 * (i + 1) + hi) * 16 + lm];
                acc1 = __builtin_amdgcn_wmma_f32_16x16x4_f32(false, a1, false, b1,
                                                             (short)0, acc1, false, false);
            }
        }

        // ---- epilogue: leaky tanh + threshold; C/D layout: M = r + 8*hi, N = lm
        #pragma unroll
        for (int r = 0; r < 8; ++r) {
            int m = m0 + r + 8 * hi;
            size_t idx = ((size_t)m * SEQ + t) * DOUTC + n0 + lm;
            float pre  = acc0[r] + acc1[r];
            float prev = (t > 0) ? out[idx - DOUTC] : 0.f;
            float s = 0.9f * prev + 0.1f * tanhf(pre);
            s = (s > 0.5f) ? (s - 0.5f) : s;
            out[idx] = s;
        }

        // ---- device-wide barrier between steps (monotonic counter in d_ws)
        __threadfence();
        __syncthreads();
        if (tid == 0) {
            __hip_atomic_fetch_add(bar, 1u, __ATOMIC_ACQ_REL, __HIP_MEMORY_SCOPE_AGENT);
            const unsigned target = (unsigned)(t + 1) * NWG;
            while (__hip_atomic_load(bar, __ATOMIC_ACQUIRE, __HIP_MEMORY_SCOPE_AGENT) < target) {
                __builtin_amdgcn_s_sleep(2);
            }
        }
        __syncthreads();
        __threadfence();
    }
}

extern "C" void kernel_launch(void* const* d_in, const int* in_sizes, int n_in,
                              void* d_out, int out_size, void* d_ws, size_t ws_size,
                              hipStream_t stream) {
    const float* x    = (const float*)d_in[0];  // inputs            (128,256,512)
    const float* wres = (const float*)d_in[1];  // reservoir_weights (2048,2048)
    const float* win  = (const float*)d_in[2];  // input_weights     (2048,512)
    float* out        = (float*)d_out;          // (128,256,2048)
    unsigned* bar     = (unsigned*)d_ws;        // 4-byte barrier counter

    (void)in_sizes; (void)n_in; (void)out_size; (void)ws_size;

    reservoir_init_kernel<<<1024, 256, 0, stream>>>(out, bar);

    const size_t lds_bytes = 8192 * sizeof(v2f) + 4096 * sizeof(v2f); // 96 KB
    reservoir_step_kernel<<<NWG, TPB, lds_bytes, stream>>>(x, wres, win, out, bar);
}